// MonotonicJoinerV1_83133386981839
// MI455X (gfx1250) — compile-verified
//
#include <hip/hip_runtime.h>
#include <hip/hip_bf16.h>

// Problem constants (match the reference).
#define B_ 8
#define T_ 256
#define U_ 64
#define D_ 640
#define V_ 1024

typedef __attribute__((ext_vector_type(2))) float v2f;
typedef __attribute__((ext_vector_type(4))) float v4f;
typedef __attribute__((ext_vector_type(8))) float v8f;

// One wave (32 lanes) computes one 16x16 tile of Out = relu(A) @ Wpart^T (+bias).
//   A    : [M, D_] row-major (source or target encodings, relu applied on load)
//   W    : [V_, 2*D_] row-major; we use columns [col0, col0+D_)
//   Out  : [M, V_] row-major
// V_WMMA_F32_16X16X4_F32 fragment layout (ISA 7.12.2):
//   A 16x4 : lane l in 0..15 -> M=l, VGPR0=K, VGPR1=K+1 ; lanes 16..31 -> K+2,K+3
//   B 4x16 : loaded column-major == rows of W (B^T), same half-wave K split
//   C/D    : VGPR r, lanes 0..15 -> (M=r, N=lane); lanes 16..31 -> (M=r+8, N=lane-16)
__global__ __launch_bounds__(32) void joiner_gemm_wmma(
    const float* __restrict__ A,
    const float* __restrict__ W,
    const float* __restrict__ bias,   // may be null
    float* __restrict__ Out,
    int col0)
{
    const int lane = threadIdx.x & 31;
    const int half = lane >> 4;       // 0: K,K+1  1: K+2,K+3
    const int l    = lane & 15;

    const int ntn = V_ / 16;
    const int tm  = blockIdx.x / ntn;
    const int tn  = blockIdx.x % ntn;
    const int m0  = tm * 16;
    const int n0  = tn * 16;

    const float* __restrict__ arow = A + (size_t)(m0 + l) * D_;
    const float* __restrict__ wrow = W + (size_t)(n0 + l) * (2 * D_) + col0;

    v8f acc = {};
    #pragma unroll 4
    for (int k = 0; k < D_; k += 4) {
        const int ka = k + 2 * half;
        const v2f av = *(const v2f*)(arow + ka);   // global_load_b64
        const v2f wv = *(const v2f*)(wrow + ka);
        v2f a; a[0] = fmaxf(av[0], 0.0f); a[1] = fmaxf(av[1], 0.0f);  // ReLU on A only
        v2f b; b[0] = wv[0];              b[1] = wv[1];
        acc = __builtin_amdgcn_wmma_f32_16x16x4_f32(
            /*neg_a=*/false, a, /*neg_b=*/false, b,
            /*c_mod=*/(short)0, acc, /*reuse_a=*/false, /*reuse_b=*/false);
    }

    const float bv = bias ? bias[n0 + l] : 0.0f;
    #pragma unroll
    for (int r = 0; r < 8; ++r) {
        Out[(size_t)(m0 + r + 8 * half) * V_ + (n0 + l)] = acc[r] + bv;
    }
}

// out[((b*T+t)*U+u)*V + v] = E[(b*T+t)*V + v] + P[(b*U+u)*V + v]
// One block per (b,t); E row pinned in registers, P rows hit L2 (P = 2MB total),
// 512MB of output streamed with non-temporal stores.
__global__ __launch_bounds__(256) void joiner_combine(
    const float* __restrict__ E,
    const float* __restrict__ P,
    float* __restrict__ Out)
{
    const int bt = blockIdx.x;          // 0 .. B_*T_-1
    const int b  = bt / T_;
    const int i  = threadIdx.x;         // 0..255 ; V_/4 == 256 float4 per row

    const v4f ev = *(const v4f*)(E + (size_t)bt * V_ + (size_t)i * 4);
    const v4f* __restrict__ prow = (const v4f*)(P + (size_t)b * U_ * V_);
    v4f* __restrict__ orow = (v4f*)Out + (size_t)bt * U_ * (V_ / 4);

    #pragma unroll 4
    for (int u = 0; u < U_; ++u) {
        const v4f pv = prow[(size_t)u * (V_ / 4) + i];
        const v4f o = ev + pv;
        __builtin_nontemporal_store(o, orow + (size_t)u * (V_ / 4) + i);
    }
}

// Bit-copy the int32 length vectors into the tail of d_out (tuple outputs 2 & 3).
__global__ void joiner_copy_lengths(const int* __restrict__ sl,
                                    const int* __restrict__ tl,
                                    int* __restrict__ out_tail)
{
    const int i = threadIdx.x;
    if (i < B_)            out_tail[i] = sl[i];
    else if (i < 2 * B_)   out_tail[i] = tl[i - B_];
}

extern "C" void kernel_launch(void* const* d_in, const int* in_sizes, int n_in,
                              void* d_out, int out_size, void* d_ws, size_t ws_size,
                              hipStream_t stream) {
    (void)in_sizes; (void)n_in; (void)out_size; (void)ws_size;

    const float* src  = (const float*)d_in[0];  // [B,T,D]
    const int*   slen = (const int*)  d_in[1];  // [B]
    const float* tgt  = (const float*)d_in[2];  // [B,U,D]
    const int*   tlen = (const int*)  d_in[3];  // [B]
    const float* W    = (const float*)d_in[4];  // [V, 2D]
    const float* bias = (const float*)d_in[5];  // [V]

    float* out = (float*)d_out;
    float* E = (float*)d_ws;                    // [B*T, V]  = 8 MB
    float* P = E + (size_t)B_ * T_ * V_;        // [B*U, V]  = 2 MB (bias folded in)

    // E = relu(src) @ W[:, :D].T
    joiner_gemm_wmma<<<(B_ * T_ / 16) * (V_ / 16), 32, 0, stream>>>(src, W, nullptr, E, 0);
    // P = relu(tgt) @ W[:, D:].T + bias
    joiner_gemm_wmma<<<(B_ * U_ / 16) * (V_ / 16), 32, 0, stream>>>(tgt, W, bias, P, D_);
    // out = broadcast-add over the (T,U) lattice
    joiner_combine<<<B_ * T_, 256, 0, stream>>>(E, P, out);
    // tuple tail: source_lengths, target_lengths
    joiner_copy_lengths<<<1, 32, 0, stream>>>(slen, tlen,
                                              (int*)(out + (size_t)B_ * T_ * U_ * V_));
}